// LSTMModel_4131758539008
// MI455X (gfx1250) — compile-verified
//
#include <hip/hip_runtime.h>

typedef __attribute__((ext_vector_type(2))) float v2f;
typedef __attribute__((ext_vector_type(8))) float v8f;

#define B_TOT 4096
#define T_    512
#define IN_   2
#define H_    12
#define OUT_  2

__device__ __forceinline__ v8f wmma4(v2f a, v2f b, v8f c) {
  // D = A(16x4 f32) * B(4x16 f32) + C(16x16 f32)
  return __builtin_amdgcn_wmma_f32_16x16x4_f32(false, a, false, b, (short)0, c,
                                               false, false);
}

// lane <-> lane^16 exchange as pure VALU (v_permlanex16_b32), avoids LDS.
__device__ __forceinline__ float xhalf(float v) {
  int i = __builtin_bit_cast(int, v);
  int r = __builtin_amdgcn_permlanex16(i, i, 0x76543210, 0xfedcba98, false, false);
  return __builtin_bit_cast(float, r);
}

// CDNA5 hardware tanh transcendental (V_TANH_F32).
#if defined(__HIP_DEVICE_COMPILE__) && __has_builtin(__builtin_amdgcn_tanhf)
__device__ __forceinline__ float tanh_fast(float x) {
  return __builtin_amdgcn_tanhf(x);
}
__device__ __forceinline__ float sigm(float x) {
  // sigmoid(x) = 0.5 + 0.5*tanh(0.5*x)
  return __builtin_fmaf(0.5f, __builtin_amdgcn_tanhf(0.5f * x), 0.5f);
}
#else
__device__ __forceinline__ float sigm(float x) {
  return __builtin_amdgcn_rcpf(1.0f + __expf(-x));
}
__device__ __forceinline__ float tanh_fast(float x) {
  return 2.0f * __builtin_amdgcn_rcpf(1.0f + __expf(-2.0f * x)) - 1.0f;
}
#endif

// h-state kept as three ready-to-use B fragments (4x16 each, K rows 4k..4k+3).
struct BFrags {
  v2f b[3];
};

// Convert a 16x16 C/D-layout h (rows: lanes<16 -> 0..7, lanes>=16 -> 8..15)
// into B fragments. B layout: vgpr0 = rows {4k | 4k+2}, vgpr1 = {4k+1 | 4k+3}.
__device__ __forceinline__ BFrags fragize(const v8f& h, bool lo) {
  BFrags f;
  float x0 = xhalf(h[0]), x1 = xhalf(h[1]);
  float x2 = xhalf(h[2]), x3 = xhalf(h[3]);
  float x6 = xhalf(h[6]), x7 = xhalf(h[7]);
  f.b[0].x = lo ? h[0] : x2;
  f.b[0].y = lo ? h[1] : x3;
  f.b[1].x = lo ? h[4] : x6;
  f.b[1].y = lo ? h[5] : x7;
  f.b[2].x = lo ? x0 : h[2];
  f.b[2].y = lo ? x1 : h[3];
  return f;
}

// g[q] += A[q][k] @ f.b[k] for k=0..2 (pure accumulate)
__device__ __forceinline__ void accum3(v8f g[4], const v2f (&a)[4][3],
                                       const BFrags& f) {
#pragma unroll
  for (int k = 0; k < 3; ++k) {
    g[0] = wmma4(a[0][k], f.b[k], g[0]);
    g[1] = wmma4(a[1][k], f.b[k], g[1]);
    g[2] = wmma4(a[2][k], f.b[k], g[2]);
    g[3] = wmma4(a[3][k], f.b[k], g[3]);
  }
}

__device__ __forceinline__ void cell_update(const v8f g[4], v8f& ht, v8f& ct) {
#pragma unroll
  for (int r = 0; r < 8; ++r) {
    float iv = sigm(g[0][r]);
    float fv = sigm(g[1][r]);
    float gv = tanh_fast(g[2][r]);
    float ov = sigm(g[3][r]);
    float c  = fv * ct[r] + iv * gv;
    ct[r] = c;
    ht[r] = ov * tanh_fast(c);
  }
}

__global__ __launch_bounds__(32) void lstm2_wmma_kernel(
    const float* __restrict__ x,
    const float* __restrict__ W_ih0, const float* __restrict__ W_hh0,
    const float* __restrict__ b_ih0, const float* __restrict__ b_hh0,
    const float* __restrict__ W_ih1, const float* __restrict__ W_hh1,
    const float* __restrict__ b_ih1, const float* __restrict__ b_hh1,
    const float* __restrict__ fc_W,  const float* __restrict__ fc_b,
    float* __restrict__ out) {
  const int lane  = threadIdx.x;   // 0..31, wave32
  const int m     = lane & 15;     // gate-row within tile / batch column
  const bool lo   = lane < 16;
  const int khalf = lo ? 0 : 2;    // A-layout: lanes>=16 carry K=+2,+3
  const int batch = blockIdx.x * 16 + m;

  // ---- persistent weight fragments (A matrices, gates x K) ----
  v2f a_hh0[4][3], a_hh1[4][3], a_ih1[4][3], a_x0[4];
  v8f bfrag1[4];  // layer-1 bias in C/D layout (initial accumulator)
#pragma unroll
  for (int q = 0; q < 4; ++q) {
    const int  row   = q * H_ + m;       // gate row in original 48-row layout
    const bool valid = (m < H_);         // rows 12..15 of each tile = zero pad
#pragma unroll
    for (int k = 0; k < 3; ++k) {
      const int c0 = 4 * k + khalf;
      a_hh0[q][k].x = valid ? W_hh0[row * H_ + c0]     : 0.0f;
      a_hh0[q][k].y = valid ? W_hh0[row * H_ + c0 + 1] : 0.0f;
      a_hh1[q][k].x = valid ? W_hh1[row * H_ + c0]     : 0.0f;
      a_hh1[q][k].y = valid ? W_hh1[row * H_ + c0 + 1] : 0.0f;
      a_ih1[q][k].x = valid ? W_ih1[row * H_ + c0]     : 0.0f;
      a_ih1[q][k].y = valid ? W_ih1[row * H_ + c0 + 1] : 0.0f;
    }
    // layer-0 extended input weights:
    //   lanes<16 : K=0,1 -> W_ih0 columns ;  lanes>=16 : K=2 -> bias0, K=3 -> 0
    const float bias0 = valid ? (b_ih0[row] + b_hh0[row]) : 0.0f;
    if (lo) {
      a_x0[q].x = valid ? W_ih0[row * IN_ + 0] : 0.0f;
      a_x0[q].y = valid ? W_ih0[row * IN_ + 1] : 0.0f;
    } else {
      a_x0[q].x = bias0;
      a_x0[q].y = 0.0f;
    }
    // layer-1 bias fragment in C/D layout: vgpr r -> gate row (lo ? r : 8+r)
#pragma unroll
    for (int r = 0; r < 8; ++r) {
      const int gr  = (lo ? r : 8 + r);
      bfrag1[q][r] = (gr < H_) ? (b_ih1[q * H_ + gr] + b_hh1[q * H_ + gr]) : 0.0f;
    }
  }

  v8f vzero;
#pragma unroll
  for (int r = 0; r < 8; ++r) vzero[r] = 0.0f;
  v8f ht0 = vzero, ct0 = vzero, ht1 = vzero, ct1 = vzero;
  BFrags bh0 = fragize(ht0, lo);  // zero fragments
  BFrags bh1 = bh0;

  const float2* __restrict__ xb =
      (const float2*)(x + (size_t)batch * T_ * IN_);

  // One full two-layer LSTM timestep (captures state by reference).
  auto step = [&](float2 xv) {
    // layer 0: g0 = (bias0 folded in A/B, K=2) + W_ih0 @ x_t + W_hh0 @ h0_prev
    v2f bx;
    bx.x = lo ? xv.x : 1.0f;  // B rows: K=0 -> x0 ; K=2 -> 1 (bias column)
    bx.y = lo ? xv.y : 0.0f;  // K=1 -> x1 ; K=3 -> 0
    v8f g0[4];
    g0[0] = wmma4(a_x0[0], bx, vzero);  // C = inline 0
    g0[1] = wmma4(a_x0[1], bx, vzero);
    g0[2] = wmma4(a_x0[2], bx, vzero);
    g0[3] = wmma4(a_x0[3], bx, vzero);
    accum3(g0, a_hh0, bh0);
    cell_update(g0, ht0, ct0);
    bh0 = fragize(ht0, lo);  // reused: layer-1 input now, recurrence next t

    // layer 1: g1 = bias1 (C operand) + W_ih1 @ h0_new + W_hh1 @ h1_prev
    v8f g1[4];
    g1[0] = wmma4(a_ih1[0][0], bh0.b[0], bfrag1[0]);
    g1[1] = wmma4(a_ih1[1][0], bh0.b[0], bfrag1[1]);
    g1[2] = wmma4(a_ih1[2][0], bh0.b[0], bfrag1[2]);
    g1[3] = wmma4(a_ih1[3][0], bh0.b[0], bfrag1[3]);
#pragma unroll
    for (int k = 1; k < 3; ++k) {
      g1[0] = wmma4(a_ih1[0][k], bh0.b[k], g1[0]);
      g1[1] = wmma4(a_ih1[1][k], bh0.b[k], g1[1]);
      g1[2] = wmma4(a_ih1[2][k], bh0.b[k], g1[2]);
      g1[3] = wmma4(a_ih1[3][k], bh0.b[k], g1[3]);
    }
    accum3(g1, a_hh1, bh1);
    cell_update(g1, ht1, ct1);
    bh1 = fragize(ht1, lo);
  };

  // Software-pipelined x stream; last iteration peeled (no clamp arithmetic).
  float2 xv = xb[0];
#pragma unroll 1
  for (int t = 0; t < T_ - 1; ++t) {
    float2 xnext = xb[t + 1];  // plain strength-reduced pointer advance
    step(xv);
    xv = xnext;
  }
  step(xv);

  // ---- FC on last hidden of layer 1: out[b][o] = sum_h h[h][b]*fcW[o][h]+fcb
  float p0 = 0.0f, p1 = 0.0f;
#pragma unroll
  for (int r = 0; r < 8; ++r) {
    const int hidx = lo ? r : 8 + r;
    const float w0 = (hidx < H_) ? fc_W[0 * H_ + hidx] : 0.0f;
    const float w1 = (hidx < H_) ? fc_W[1 * H_ + hidx] : 0.0f;
    p0 += ht1[r] * w0;
    p1 += ht1[r] * w1;
  }
  p0 += xhalf(p0);  // combine lower(rows 0-7) + upper(rows 8-11)
  p1 += xhalf(p1);
  if (lo) {
    out[batch * OUT_ + 0] = p0 + fc_b[0];
    out[batch * OUT_ + 1] = p1 + fc_b[1];
  }
}

extern "C" void kernel_launch(void* const* d_in, const int* in_sizes, int n_in,
                              void* d_out, int out_size, void* d_ws,
                              size_t ws_size, hipStream_t stream) {
  const float* x     = (const float*)d_in[0];
  const float* W_ih0 = (const float*)d_in[1];
  const float* W_hh0 = (const float*)d_in[2];
  const float* b_ih0 = (const float*)d_in[3];
  const float* b_hh0 = (const float*)d_in[4];
  const float* W_ih1 = (const float*)d_in[5];
  const float* W_hh1 = (const float*)d_in[6];
  const float* b_ih1 = (const float*)d_in[7];
  const float* b_hh1 = (const float*)d_in[8];
  const float* fc_W  = (const float*)d_in[9];
  const float* fc_b  = (const float*)d_in[10];
  float* out = (float*)d_out;

  dim3 grid(B_TOT / 16);  // 256 independent recurrence chains, one wave each
  dim3 block(32);
  hipLaunchKernelGGL(lstm2_wmma_kernel, grid, block, 0, stream, x, W_ih0, W_hh0,
                     b_ih0, b_hh0, W_ih1, W_hh1, b_ih1, b_hh1, fc_W, fc_b, out);
}